// EKF_v2_89464168776308
// MI455X (gfx1250) — compile-verified
//
#include <hip/hip_runtime.h>
#include <hip/hip_bf16.h>

typedef float v2f __attribute__((ext_vector_type(2)));
typedef float v8f __attribute__((ext_vector_type(8)));

// ---------------------------------------------------------------------------
// Wave32 sum reduction (deterministic butterfly)
// ---------------------------------------------------------------------------
__device__ __forceinline__ float wave_sum(float v) {
#pragma unroll
    for (int off = 16; off > 0; off >>= 1)
        v += __shfl_xor(v, off, 32);
    return v;
}

// ---------------------------------------------------------------------------
// Kernel 1: per-batch observation-noise diag via WMMA f32 16x16x4
//   H = relu(E(16x128) @ nw1(128x32) + nb1); d = H @ nw2(32x2) + nb2
//   diag = (d + lob)^2 + fob        -> d_ws[B*2]
// One wave (32 threads) handles a tile of 16 batches.
// ---------------------------------------------------------------------------
__global__ __launch_bounds__(32) void enkf_noise_diag(
    const float* __restrict__ enc,    // B x 128
    const float* __restrict__ nw1,    // 128 x 32
    const float* __restrict__ nb1,    // 32
    const float* __restrict__ nw2,    // 32 x 2
    const float* __restrict__ nb2,    // 2
    const float* __restrict__ lob,    // 2
    const float* __restrict__ fob,    // 2
    float* __restrict__ diag_out)     // B x 2
{
    __shared__ float lds_e[16][128];  // encoding tile
    __shared__ float lds_h[16][32];   // hidden activations

    const int lane = threadIdx.x;     // 0..31
    const int b0   = blockIdx.x * 16; // first batch of tile

    // Coalesced float4 stage of 16 encoding rows into LDS
#pragma unroll 4
    for (int r = 0; r < 16; ++r) {
        const float4 v = ((const float4*)(enc + (size_t)(b0 + r) * 128))[lane];
        ((float4*)&lds_e[r][0])[lane] = v;
    }
    __syncthreads();

    const int m     = lane & 15;        // A-matrix row (batch in tile)
    const int khalf = (lane >> 4) * 2;  // K sub-pair selected by lane[4]

#pragma unroll
    for (int nt = 0; nt < 2; ++nt) {
        const int col = nt * 16 + (lane & 15);   // B/C column
        v8f c = {};
#pragma unroll
        for (int kt = 0; kt < 32; ++kt) {        // K = 128 in steps of 4
            const int k0 = kt * 4 + khalf;
            v2f a, bm;
            a.x  = lds_e[m][k0];
            a.y  = lds_e[m][k0 + 1];
            bm.x = nw1[(size_t)k0 * 32 + col];
            bm.y = nw1[(size_t)(k0 + 1) * 32 + col];
            c = __builtin_amdgcn_wmma_f32_16x16x4_f32(
                    false, a, false, bm, (short)0, c, false, false);
        }
        // C layout: VGPR r -> row r (lanes 0-15) or r+8 (lanes 16-31)
        const float bias = nb1[col];
        const int   r0   = (lane >> 4) * 8;
#pragma unroll
        for (int r = 0; r < 8; ++r) {
            const float h = c[r] + bias;
            lds_h[r0 + r][col] = fmaxf(h, 0.0f);
        }
    }
    __syncthreads();

    // Layer 2: 32 outputs = 16 batches x 2 obs dims, one per lane
    {
        const int mb = lane & 15;
        const int z  = lane >> 4;
        float acc = nb2[z];
#pragma unroll
        for (int c2 = 0; c2 < 32; ++c2)
            acc = fmaf(lds_h[mb][c2], nw2[c2 * 2 + z], acc);
        const float dv = acc + lob[z];
        diag_out[(size_t)(b0 + mb) * 2 + z] = fmaf(dv, dv, fob[z]);
    }
}

// ---------------------------------------------------------------------------
// Kernel 2: full EnKF update. One block (128 threads = 4 waves) per batch;
// thread t = ensemble member t. The 16->32 MLP layer runs on the fp32 WMMA
// pipe (per wave: M=32, K=16, N=32 -> 16 wmma ops, no padding waste).
// ---------------------------------------------------------------------------
__global__ __launch_bounds__(128) void enkf_update(
    const float* __restrict__ state_old, // B x 128 x 5
    const float* __restrict__ yobs,      // B x 128 x 2
    const float* __restrict__ pw1,       // 2 x 16
    const float* __restrict__ pb1,       // 16
    const float* __restrict__ pw2,       // 16 x 32
    const float* __restrict__ pb2,       // 32
    const float* __restrict__ pw3,       // 32 x 2
    const float* __restrict__ pb3,       // 2
    const float* __restrict__ rdiag,     // B x 2
    float* __restrict__ xnew)            // B x 128 x 5
{
    __shared__ float lds_h1[128][16];    // 8 KB
    __shared__ float lds_h2[128][32];    // 16 KB
    __shared__ float lds_part[4][16];    // cross-wave reduction partials

    const int t = threadIdx.x;           // member 0..127
    const int b = blockIdx.x;
    const size_t base = ((size_t)b * 128 + t) * 5;

    // ---- process-model kinematics --------------------------------------
    const float s0 = state_old[base + 0];
    const float s1 = state_old[base + 1];
    const float s2 = state_old[base + 2];
    const float s3 = state_old[base + 3];
    const float s4 = state_old[base + 4];
    const float theta = s2 + s4;
    const float v  = s3;
    const float td = s4;
    const float xx = fmaf(v, __sinf(theta), s0);
    const float yy = fmaf(v, __cosf(theta), s1);

    // ---- MLP layer 1 (2 -> 16), VALU, stage h1 to LDS ------------------
#pragma unroll
    for (int j = 0; j < 16; ++j) {
        const float h = fmaf(v, pw1[j], fmaf(td, pw1[16 + j], pb1[j]));
        lds_h1[t][j] = fmaxf(h, 0.0f);
    }
    __syncthreads();

    // ---- MLP layer 2 (16 -> 32) via WMMA f32 16x16x4 -------------------
    const int wave  = t >> 5;
    const int lane  = t & 31;
    const int mrow  = lane & 15;
    const int khalf = (lane >> 4) * 2;
#pragma unroll
    for (int mt = 0; mt < 2; ++mt) {
        const int mb = wave * 32 + mt * 16;      // 16-particle tile
#pragma unroll
        for (int nt = 0; nt < 2; ++nt) {
            const int col = nt * 16 + (lane & 15);
            v8f c = {};
#pragma unroll
            for (int kt = 0; kt < 4; ++kt) {     // K = 16 in steps of 4
                const int k0 = kt * 4 + khalf;
                v2f a, bm;
                a.x  = lds_h1[mb + mrow][k0];
                a.y  = lds_h1[mb + mrow][k0 + 1];
                bm.x = pw2[k0 * 32 + col];
                bm.y = pw2[(k0 + 1) * 32 + col];
                c = __builtin_amdgcn_wmma_f32_16x16x4_f32(
                        false, a, false, bm, (short)0, c, false, false);
            }
            const float bias = pb2[col];
            const int   r0   = (lane >> 4) * 8;
#pragma unroll
            for (int r = 0; r < 8; ++r) {
                const float h = c[r] + bias;
                lds_h2[mb + r0 + r][col] = fmaxf(h, 0.0f);
            }
        }
    }
    __syncthreads();

    // ---- MLP layer 3 (32 -> 2) + residual ------------------------------
    float o0 = v  + pb3[0];
    float o1 = td + pb3[1];
#pragma unroll
    for (int c2 = 0; c2 < 32; ++c2) {
        const float h = lds_h2[t][c2];
        o0 = fmaf(h, pw3[c2 * 2 + 0], o0);
        o1 = fmaf(h, pw3[c2 * 2 + 1], o1);
    }

    float Xp[5] = { xx, yy, theta, o0, o1 };

    // ---- ensemble means -------------------------------------------------
    {
        float s[5];
#pragma unroll
        for (int c2 = 0; c2 < 5; ++c2) s[c2] = wave_sum(Xp[c2]);
        if (lane == 0) {
#pragma unroll
            for (int c2 = 0; c2 < 5; ++c2) lds_part[wave][c2] = s[c2];
        }
    }
    __syncthreads();
    float mean[5];
#pragma unroll
    for (int c2 = 0; c2 < 5; ++c2)
        mean[c2] = (lds_part[0][c2] + lds_part[1][c2] +
                    lds_part[2][c2] + lds_part[3][c2]) * (1.0f / 128.0f);
    __syncthreads();

    // ---- covariance sums: P[x][z] = sum_n A_x * HA_z (10 values) -------
    float av[5];
#pragma unroll
    for (int c2 = 0; c2 < 5; ++c2) av[c2] = Xp[c2] - mean[c2];
    const float ha0 = av[3], ha1 = av[4];

    float P[10];
#pragma unroll
    for (int x = 0; x < 5; ++x) {
        P[x * 2 + 0] = wave_sum(av[x] * ha0);
        P[x * 2 + 1] = wave_sum(av[x] * ha1);
    }
    if (lane == 0) {
#pragma unroll
        for (int i = 0; i < 10; ++i) lds_part[wave][i] = P[i];
    }
    __syncthreads();
#pragma unroll
    for (int i = 0; i < 10; ++i)
        P[i] = lds_part[0][i] + lds_part[1][i] + lds_part[2][i] + lds_part[3][i];

    // ---- 2x2 solve: Kt = Pzz^-1 @ Pxz^T (redundant per thread) ---------
    const float dinv  = 1.0f / 127.0f;
    const float R0    = rdiag[(size_t)b * 2 + 0];
    const float R1    = rdiag[(size_t)b * 2 + 1];
    const float pzz00 = P[6] * dinv + R0;   // sum ha0*ha0
    const float pzz01 = P[7] * dinv;        // sum ha0*ha1
    const float pzz10 = P[8] * dinv;        // sum ha1*ha0
    const float pzz11 = P[9] * dinv + R1;   // sum ha1*ha1
    const float idet  = 1.0f / (pzz00 * pzz11 - pzz01 * pzz10);

    float Kt0[5], Kt1[5];
#pragma unroll
    for (int x = 0; x < 5; ++x) {
        const float px0 = P[x * 2 + 0] * dinv;   // Pxz^T[0][x]
        const float px1 = P[x * 2 + 1] * dinv;   // Pxz^T[1][x]
        Kt0[x] = (pzz11 * px0 - pzz01 * px1) * idet;
        Kt1[x] = (pzz00 * px1 - pzz10 * px0) * idet;
    }

    // ---- Kalman update --------------------------------------------------
    const size_t yb = ((size_t)b * 128 + t) * 2;
    const float in0 = yobs[yb + 0] - Xp[3];
    const float in1 = yobs[yb + 1] - Xp[4];
#pragma unroll
    for (int c2 = 0; c2 < 5; ++c2)
        xnew[base + c2] = Xp[c2] + in0 * Kt0[c2] + in1 * Kt1[c2];
}

// ---------------------------------------------------------------------------
extern "C" void kernel_launch(void* const* d_in, const int* in_sizes, int n_in,
                              void* d_out, int out_size, void* d_ws, size_t ws_size,
                              hipStream_t stream) {
    const float* state_old = (const float*)d_in[0];
    const float* yobs      = (const float*)d_in[1];
    const float* enc       = (const float*)d_in[2];
    const float* pw1       = (const float*)d_in[3];
    const float* pb1       = (const float*)d_in[4];
    const float* pw2       = (const float*)d_in[5];
    const float* pb2       = (const float*)d_in[6];
    const float* pw3       = (const float*)d_in[7];
    const float* pb3       = (const float*)d_in[8];
    const float* nw1       = (const float*)d_in[9];
    const float* nb1       = (const float*)d_in[10];
    const float* nw2       = (const float*)d_in[11];
    const float* nb2       = (const float*)d_in[12];
    const float* lob       = (const float*)d_in[13];
    const float* fob       = (const float*)d_in[14];
    float* out  = (float*)d_out;
    float* diag = (float*)d_ws;          // B*2 floats of scratch

    const int B = in_sizes[0] / (128 * 5);

    enkf_noise_diag<<<B / 16, 32, 0, stream>>>(enc, nw1, nb1, nw2, nb2,
                                               lob, fob, diag);
    enkf_update<<<B, 128, 0, stream>>>(state_old, yobs, pw1, pb1, pw2, pb2,
                                       pw3, pb3, diag, out);
}